// GraphSAGE_2963527434976
// MI455X (gfx1250) — compile-verified
//
#include <hip/hip_runtime.h>

#define N_NODES 100000
#define N_EDGES 1000000
#define D_IN 128
#define D_L2 40

typedef float v2f __attribute__((ext_vector_type(2)));
typedef float v8f __attribute__((ext_vector_type(8)));

// ---------------------------------------------------------------- utilities
__global__ void zero_kernel(float* __restrict__ p, long n) {
    long i = (long)blockIdx.x * blockDim.x + threadIdx.x;
    long stride = (long)gridDim.x * blockDim.x;
    for (; i < n; i += stride) p[i] = 0.0f;
}

__global__ void deg_kernel(const int* __restrict__ dst, float* __restrict__ deg) {
    int e = blockIdx.x * blockDim.x + threadIdx.x;
    if (e < N_EDGES) atomicAdd(&deg[dst[e]], 1.0f);
}

// one thread = one (edge, 4-float chunk); 32 chunks cover the 128-dim row
__global__ void scatter_kernel(const int* __restrict__ src, const int* __restrict__ dst,
                               const float* __restrict__ h, float* __restrict__ neigh) {
    long gid = (long)blockIdx.x * blockDim.x + threadIdx.x;
    const long total = (long)N_EDGES * (D_IN / 4);
    if (gid >= total) return;
    int e = (int)(gid >> 5);          // / 32 chunks
    int c = ((int)gid & 31) * 4;
    int s = src[e];
    int d = dst[e];
    const float4 v = *(const float4*)(h + (long)s * D_IN + c);
    float* o = neigh + (long)d * D_IN + c;
    atomicAdd(o + 0, v.x);
    atomicAdd(o + 1, v.y);
    atomicAdd(o + 2, v.z);
    atomicAdd(o + 3, v.w);
}

// agg = (neigh + h_self) / (deg + 1), computed in-place over neigh
__global__ void agg_kernel(float* __restrict__ neigh, const float* __restrict__ h,
                           const float* __restrict__ deg) {
    long i = (long)blockIdx.x * blockDim.x + threadIdx.x;
    const long total = (long)N_NODES * D_IN;
    if (i >= total) return;
    int node = (int)(i >> 7);  // / 128
    neigh[i] = (neigh[i] + h[i]) / (deg[node] + 1.0f);
}

// ------------------------------------------------------------- WMMA GEMM
// One wave32 computes one 16x16 tile of out = A[N_NODES x 128] @ W[128 x ncols] + bias.
// K-loop in steps of 4 using V_WMMA_F32_16X16X4_F32 (fp32-exact vs reference).
// VGPR layouts (ISA 7.12.2):
//   A 16x4:  lane lo=M, k = 2*hi + j   (j = vgpr index, hi = lane>>4)
//   B 4x16:  lane lo=N, k = 2*hi + j
//   C/D:     row M = r + 8*hi, col N = lo, r = vgpr 0..7
__global__ __launch_bounds__(256) void gemm_wmma_kernel(
        const float* __restrict__ A, const float* __restrict__ W,
        const float* __restrict__ bias, float* __restrict__ out,
        int ncols, int ntiles_n, int total_tiles, int do_relu) {
    int wave = threadIdx.x >> 5;              // 8 waves per block
    int tile = blockIdx.x * 8 + wave;         // wave-uniform
    if (tile >= total_tiles) return;          // uniform exit: EXEC stays all-1s below
    int lane = threadIdx.x & 31;
    int lo = lane & 15;
    int hi = lane >> 4;

    int mt = tile / ntiles_n;
    int nt = tile - mt * ntiles_n;
    int m0 = mt * 16;
    int n0 = nt * 16;
    int n  = n0 + lo;
    int nc = (n < ncols) ? n : (ncols - 1);   // clamp for safe OOB-tile loads
    bool nvalid = (n < ncols);

    const float* __restrict__ arow = A + (long)(m0 + lo) * D_IN;

    v8f c = {};
    #pragma unroll 4
    for (int kk = 0; kk < D_IN; kk += 4) {
        int k0 = kk + 2 * hi;
        v2f a, b;
        a.x = arow[k0 + 0];
        a.y = arow[k0 + 1];
        float w0 = W[(long)(k0 + 0) * ncols + nc];
        float w1 = W[(long)(k0 + 1) * ncols + nc];
        b.x = nvalid ? w0 : 0.0f;
        b.y = nvalid ? w1 : 0.0f;
        // (neg_a, A, neg_b, B, c_mod, C, reuse_a, reuse_b)
        c = __builtin_amdgcn_wmma_f32_16x16x4_f32(false, a, false, b, (short)0, c,
                                                  false, false);
    }

    float bv = bias[nc];
    if (nvalid) {
        #pragma unroll
        for (int r = 0; r < 8; ++r) {
            float v = c[r] + bv;
            if (do_relu) v = fmaxf(v, 0.0f);
            out[(long)(m0 + r + 8 * hi) * ncols + n] = v;
        }
    }
}

// ---------------------------------------------------------------- launcher
extern "C" void kernel_launch(void* const* d_in, const int* in_sizes, int n_in,
                              void* d_out, int out_size, void* d_ws, size_t ws_size,
                              hipStream_t stream) {
    const float* feats = (const float*)d_in[0];
    const int*   src   = (const int*)d_in[1];
    const int*   dst   = (const int*)d_in[2];
    const float* W1    = (const float*)d_in[3];
    const float* b1    = (const float*)d_in[4];
    const float* W2    = (const float*)d_in[5];
    const float* b2    = (const float*)d_in[6];

    float* out = (float*)d_out;
    float* h1  = out;                           // [N_NODES, 128]
    float* h2  = out + (long)N_NODES * D_IN;    // [N_NODES, 40]

    float* deg   = (float*)d_ws;                // [N_NODES]
    float* neigh = deg + N_NODES;               // [N_NODES, 128] (16B-aligned offset)

    const long feat_elems = (long)N_NODES * D_IN;

    // 1) zero deg + neigh (contiguous)
    zero_kernel<<<2048, 256, 0, stream>>>(deg, feat_elems + N_NODES);

    // 2) in-degree
    deg_kernel<<<(N_EDGES + 255) / 256, 256, 0, stream>>>(dst, deg);

    // 3) layer-1 scatter: neigh[dst] += feats[src]
    const long sc_threads = (long)N_EDGES * (D_IN / 4);
    int sc_blocks = (int)((sc_threads + 255) / 256);
    scatter_kernel<<<sc_blocks, 256, 0, stream>>>(src, dst, feats, neigh);

    // 4) agg1 in-place
    agg_kernel<<<(int)((feat_elems + 255) / 256), 256, 0, stream>>>(neigh, feats, deg);

    // 5) h1 = relu(agg1 @ W1 + b1)  [128 cols -> 8 n-tiles, 6250 m-tiles]
    {
        int ntiles_n = D_IN / 16;                        // 8
        int total = (N_NODES / 16) * ntiles_n;           // 50000
        gemm_wmma_kernel<<<(total + 7) / 8, 256, 0, stream>>>(
            neigh, W1, b1, h1, D_IN, ntiles_n, total, /*relu=*/1);
    }

    // 6) re-zero neigh only
    zero_kernel<<<2048, 256, 0, stream>>>(neigh, feat_elems);

    // 7) layer-2 scatter: neigh[dst] += h1[src]
    scatter_kernel<<<sc_blocks, 256, 0, stream>>>(src, dst, h1, neigh);

    // 8) agg2 in-place
    agg_kernel<<<(int)((feat_elems + 255) / 256), 256, 0, stream>>>(neigh, h1, deg);

    // 9) h2 = agg2 @ W2 + b2  [40 cols -> 3 n-tiles padded, guarded]
    {
        int ntiles_n = (D_L2 + 15) / 16;                 // 3
        int total = (N_NODES / 16) * ntiles_n;           // 18750
        gemm_wmma_kernel<<<(total + 7) / 8, 256, 0, stream>>>(
            neigh, W2, b2, h2, D_L2, ntiles_n, total, /*relu=*/0);
    }
}